// TypeRWSEEdgeEncoder_46720654246114
// MI455X (gfx1250) — compile-verified
//
#include <hip/hip_runtime.h>
#include <hip/hip_bf16.h>

typedef __attribute__((ext_vector_type(2))) float v2f;
typedef __attribute__((ext_vector_type(8))) float v8f;

#define B_    8
#define N_    512
#define K_    16
#define EH_   32      // emb_dim/2
#define NT_   5       // num edge types
#define E_    32768
#define CELLS (B_ * N_ * N_)   // 2,097,152 dense adjacency cells

// ---------------------------------------------------------------------------
// Kernel 0: zero the dense adjacency accumulator A (int32 per cell) in d_ws.
// ---------------------------------------------------------------------------
__global__ __launch_bounds__(256) void zeroA_kernel(int4* __restrict__ A4) {
    const int i = blockIdx.x * 256 + threadIdx.x;   // CELLS/4 int4's
    A4[i] = make_int4(0, 0, 0, 0);
}

// ---------------------------------------------------------------------------
// Kernel 1: scatter-add (edge_attr + 1) into A at (g, local_src, local_dst).
// .add semantics of the reference => atomicAdd (duplicate edges accumulate).
// ---------------------------------------------------------------------------
__global__ __launch_bounds__(256) void scatter_kernel(const int* __restrict__ edge_index,
                                                      const int* __restrict__ edge_attr,
                                                      int* __restrict__ A) {
    const int e = blockIdx.x * 256 + threadIdx.x;
    if (e >= E_) return;
    const int s = edge_index[e];        // row 0: src (global node id)
    const int d = edge_index[E_ + e];   // row 1: dst
    const int g  = s / N_;              // graph id (ordered node_batch)
    const int li = s % N_;              // local indices (equal-sized graphs)
    const int lj = d % N_;
    atomicAdd(&A[(g * N_ + li) * N_ + lj], edge_attr[e] + 1);
}

// ---------------------------------------------------------------------------
// Kernel 2: fused dense-output kernel.
//   - pe half  : edge_RWSE[tile(16 rows) x 16] @ W[16 x 32] + b via
//                V_WMMA_F32_16X16X4_F32 (2 N-blocks x 4 K-chunks per tile).
//   - type half: emb_table[clamp(A + diag, 0, 4)] gather.
// One wave per 16-cell tile; 8 waves / 256-thread block (wave32).
//
// f32 WMMA fragment layouts (ISA 7.12.2, wave32):
//   A 16x4 : lane<16 -> M=lane, {v0,v1}={K0,K1}; lane>=16 -> {K2,K3}
//   B 4x16 : lane<16 -> N=lane, {v0,v1}={K0,K1}; lane>=16 -> {K2,K3}
//   C/D    : vgpr v, element (M = v + 8*(lane>>4), N = lane&15)
// ---------------------------------------------------------------------------
__global__ __launch_bounds__(256) void fused_dense_kernel(const float* __restrict__ rwse,   // [CELLS,16]
                                                          const float* __restrict__ emb,    // [5,32]
                                                          const float* __restrict__ W,      // [16,32]
                                                          const float* __restrict__ bias,   // [32]
                                                          const int*   __restrict__ A,      // [CELLS]
                                                          float*       __restrict__ out) {  // [CELLS,64]
    const int lane = threadIdx.x & 31;
    const int wave = threadIdx.x >> 5;
    const long tile = (long)blockIdx.x * 8 + wave;
    const long r0   = tile * 16;                 // first cell of this wave's tile

    const int m  = lane & 15;                    // M (for A frag) / N (for B frag)
    const int kh = lane >> 4;                    // which K-pair this lane holds

    // ---- A-matrix fragments: 4 chunks of K=4 from 16 rows of edge_RWSE ----
    v2f afrag[4];
    {
        const float* rowp = rwse + (r0 + m) * K_;
#pragma unroll
        for (int kc = 0; kc < 4; ++kc) {
            const float* p = rowp + kc * 4 + kh * 2;
            afrag[kc].x = p[0];
            afrag[kc].y = p[1];
        }
    }

    // ---- B-matrix fragments from W (row-major [16][32]) ----
    v2f bfrag[2][4];
#pragma unroll
    for (int nb = 0; nb < 2; ++nb) {
#pragma unroll
        for (int kc = 0; kc < 4; ++kc) {
            const int krow = kc * 4 + kh * 2;
            const int col  = nb * 16 + m;
            bfrag[nb][kc].x = W[krow * EH_ + col];
            bfrag[nb][kc].y = W[(krow + 1) * EH_ + col];
        }
    }

    // ---- GEMM: two 16x16 accumulators (columns 32..47 and 48..63 of out) ----
#pragma unroll
    for (int nb = 0; nb < 2; ++nb) {
        v8f acc = {};
#pragma unroll
        for (int kc = 0; kc < 4; ++kc) {
            acc = __builtin_amdgcn_wmma_f32_16x16x4_f32(
                /*neg_a=*/false, afrag[kc],
                /*neg_b=*/false, bfrag[nb][kc],
                /*c_mod=*/(short)0, acc,
                /*reuse_a=*/false, /*reuse_b=*/false);
        }
        const int   col = nb * 16 + m;           // N index within emb half
        const float bv  = bias[col];
#pragma unroll
        for (int v = 0; v < 8; ++v) {
            const long row = r0 + v + 8 * kh;    // cell index this element belongs to
            out[row * 64 + 32 + col] = acc[v] + bv;
        }
    }

    // ---- Type half: gather emb_table[clamp(A + diag)] into columns 0..31 ----
#pragma unroll 4
    for (int mm = 0; mm < 16; ++mm) {
        const long c = r0 + mm;
        const int  j = (int)(c % N_);
        const int  i = (int)((c / N_) % N_);
        int a = A[c] + ((i == j) ? 1 : 0);       // self-loop fill = 1 on diagonal
        a = a < 0 ? 0 : (a > NT_ - 1 ? NT_ - 1 : a);   // jax OOB-index clamp
        out[c * 64 + lane] = emb[a * EH_ + lane];
    }
}

// ---------------------------------------------------------------------------
// Kernel 3: edge_attr_emb = emb_table[edge_attr]  -> out[CELLS*64 ...], [E,32]
// ---------------------------------------------------------------------------
__global__ __launch_bounds__(256) void attr_emb_kernel(const int* __restrict__ edge_attr,
                                                       const float* __restrict__ emb,
                                                       float* __restrict__ out2) {
    const int idx = blockIdx.x * 256 + threadIdx.x;   // E*32 elements
    const int e   = idx >> 5;
    const int col = idx & 31;
    int a = edge_attr[e];
    a = a < 0 ? 0 : (a > NT_ - 1 ? NT_ - 1 : a);
    out2[idx] = emb[a * EH_ + col];
}

// ---------------------------------------------------------------------------
extern "C" void kernel_launch(void* const* d_in, const int* in_sizes, int n_in,
                              void* d_out, int out_size, void* d_ws, size_t ws_size,
                              hipStream_t stream) {
    const float* rwse      = (const float*)d_in[0];   // [B*N*N, 16]
    const float* emb_table = (const float*)d_in[1];   // [5, 32]
    const float* W         = (const float*)d_in[2];   // [16, 32]
    const float* bias      = (const float*)d_in[3];   // [32]
    const int*   edge_idx  = (const int*)d_in[4];     // [2, E]
    const int*   edge_attr = (const int*)d_in[5];     // [E]
    // d_in[6] node_batch, d_in[7] B, d_in[8] N_max: implied by static shapes.

    float* out  = (float*)d_out;                      // [CELLS,64] then [E,32]
    int*   A    = (int*)d_ws;                         // CELLS int32 = 8 MB scratch

    // 0) zero adjacency accumulator (CELLS/4 int4 stores)
    zeroA_kernel<<<CELLS / 4 / 256, 256, 0, stream>>>((int4*)A);
    // 1) scatter-add edge types
    scatter_kernel<<<(E_ + 255) / 256, 256, 0, stream>>>(edge_idx, edge_attr, A);
    // 2) fused WMMA GEMM + type embedding (CELLS/16 tiles, 8 waves/block)
    fused_dense_kernel<<<CELLS / 16 / 8, 256, 0, stream>>>(rwse, emb_table, W, bias, A, out);
    // 3) edge_attr embedding
    attr_emb_kernel<<<(E_ * EH_) / 256, 256, 0, stream>>>(edge_attr, emb_table,
                                                          out + (long)CELLS * 64);
}